// ODFNet_semseg_21285857919073
// MI455X (gfx1250) — compile-verified
//
#include <hip/hip_runtime.h>
#include <hip/hip_bf16.h>

typedef __attribute__((ext_vector_type(2))) float v2f;
typedef __attribute__((ext_vector_type(8))) float v8f;
typedef __attribute__((ext_vector_type(4))) unsigned int u32x4;
typedef __attribute__((ext_vector_type(4))) int i32x4;
typedef __attribute__((ext_vector_type(8))) int i32x8;

#define BATCH 4
#define NPTS  4096
#define ROWS  16          // rows per block (one WMMA M tile)
#define KNN   33
#define NV    42
#define FINF  3.4028235e38f
#define PTS_DW (NPTS * 3)                 // 12288 dwords = 48 KB per batch

// ---------------------------------------------------------------------------
// Kernel 1: fused pairwise-distance + top-33 selection.
// Block = 256 threads (8 waves), handles 16 rows of one batch.
//  - TDM (tensor_load_to_lds) stages the batch's 48 KB of points into LDS
//    once per block, tracked by TENSORcnt (async, overlapped with A setup).
//  - Phase 1: waves split the 256 column tiles; each wave computes 16x16
//    FP32 Gram tiles with V_WMMA_F32_16X16X4_F32 (unroll 2 -> two chains in
//    flight) and writes |r|^2 - 2<r,c> + |c|^2 into a 16x4096 LDS row-block.
//  - Phase 2: wave w extracts the 33 smallest for rows 2w, 2w+1 via repeated
//    wave32 argmin (shfl_xor butterfly) + mark-with-INF, ordered with
//    s_wait_dscnt.
// LDS: 256 KB dist + 48 KB points + 64 B norms = ~304 KB <= 320 KB/WGP.
// ---------------------------------------------------------------------------
__global__ __launch_bounds__(256) void knn_kernel(const float* __restrict__ pts,
                                                  int* __restrict__ nn_idx) {
  extern __shared__ float smem[];
  float* dist    = smem;                        // ROWS*NPTS
  float* pts_lds = smem + ROWS * NPTS;          // PTS_DW
  float* sqr     = pts_lds + PTS_DW;            // ROWS

  const int b    = blockIdx.x / (NPTS / ROWS);
  const int rt   = blockIdx.x % (NPTS / ROWS);
  const int row0 = rt * ROWS;
  const int lane = threadIdx.x & 31;
  const int wave = threadIdx.x >> 5;            // 0..7
  const int half = lane >> 4;                   // lane halves per ISA layout
  const int l16  = lane & 15;
  const float* P = pts + (size_t)b * NPTS * 3;

  // --- TDM: async DMA of this batch's points (48 KB) into LDS -------------
  if (wave == 0) {
    const unsigned long long ga = (unsigned long long)(uintptr_t)P;
    const unsigned lds_a = (unsigned)(uintptr_t)pts_lds;   // group-relative
    // D# group0: count=1, lds_addr, global_addr[56:0], type=2 ("image")
    u32x4 g0 = { 0x1u,
                 lds_a,
                 (unsigned)(ga & 0xFFFFFFFFull),
                 ((unsigned)(ga >> 32) & 0x01FFFFFFu) | 0x80000000u };
    // D# group1: mask=0, data_size=4B, tensor_dim0=tile_dim0=12288,
    //            tensor_dim1=tile_dim1=1, stride0=12288
    i32x8 g1 = { 0x00020000, 0x30000000, 0x00010000, 0x30000000,
                 0x00000001, 0x00003000, 0, 0 };
    i32x4 gz = { 0, 0, 0, 0 };
#if __clang_major__ >= 23
    i32x8 gz8 = { 0, 0, 0, 0, 0, 0, 0, 0 };
    __builtin_amdgcn_tensor_load_to_lds(g0, g1, gz, gz, gz8, 0);
#else
    __builtin_amdgcn_tensor_load_to_lds(g0, g1, gz, gz, 0);
#endif
  }

  // --- A tile: rows row0..row0+15, K = {x, y, z, 0} ------------------------
  // ISA 32-bit A 16x4 layout: VGPR0 = K0 (lanes 0-15) / K2 (lanes 16-31),
  //                           VGPR1 = K1 (lanes 0-15) / K3 (lanes 16-31).
  const float rx = P[(row0 + l16) * 3 + 0];
  const float ry = P[(row0 + l16) * 3 + 1];
  const float rz = P[(row0 + l16) * 3 + 2];
  if (wave == 0 && half == 0) sqr[l16] = rx * rx + ry * ry + rz * rz;
  v2f a;
  a.x = half ? rz : rx;
  a.y = half ? 0.0f : ry;

  if (wave == 0) __builtin_amdgcn_s_wait_tensorcnt(0);   // DMA landed
  __syncthreads();

  // Row norms needed by this lane's 8 C-matrix slots (M = g + 8*half).
  float srow[8];
#pragma unroll
  for (int g = 0; g < 8; ++g) srow[g] = sqr[g + 8 * half];

  // --- Phase 1: WMMA distance tiles (B tiles come from LDS) ----------------
#pragma unroll 2
  for (int j = wave; j < NPTS / 16; j += 8) {
    const int col0 = j * 16;
    const float cx = pts_lds[(col0 + l16) * 3 + 0];
    const float cy = pts_lds[(col0 + l16) * 3 + 1];
    const float cz = pts_lds[(col0 + l16) * 3 + 2];
    const float csq = cx * cx + cy * cy + cz * cz;
    v2f bmat;
    bmat.x = half ? cz : cx;
    bmat.y = half ? 0.0f : cy;

    v8f c = {};
    // D = A(16x4) x B(4x16) + C  -> 16x16 FP32 inner products <p_row, p_col>
    c = __builtin_amdgcn_wmma_f32_16x16x4_f32(false, a, false, bmat,
                                              (short)0, c, false, false);
#pragma unroll
    for (int g = 0; g < 8; ++g) {
      const int m = g + 8 * half;               // C layout: VGPR g -> M=g / M=g+8
      dist[m * NPTS + col0 + l16] = srow[g] - 2.0f * c[g] + csq;
    }
  }
  __syncthreads();

  // --- Phase 2: top-33 per row --------------------------------------------
  for (int rr = 0; rr < 2; ++rr) {
    const int r = wave * 2 + rr;
    float* drow = &dist[r * NPTS];
    const int out_base = ((b * NPTS) + row0 + r) * KNN;
    for (int it = 0; it < KNN; ++it) {
      float bm = FINF;
      int   bi = NPTS;
      for (int cc = lane; cc < NPTS; cc += 32) {
        const float v = drow[cc];
        if (v < bm || (v == bm && cc < bi)) { bm = v; bi = cc; }
      }
#pragma unroll
      for (int off = 16; off; off >>= 1) {
        const float om = __shfl_xor(bm, off);
        const int   oi = __shfl_xor(bi, off);
        if (om < bm || (om == bm && oi < bi)) { bm = om; bi = oi; }
      }
      if (lane == 0) {
        nn_idx[out_base + it] = bi;
        drow[bi] = FINF;                        // remove winner
      }
      // order lane0's LDS store before next scan (wave-internal, split cnt)
      asm volatile("s_wait_dscnt 0" ::: "memory");
    }
  }
}

// ---------------------------------------------------------------------------
// Kernel 2: per-point ODF + color diff vectors. One wave per point,
// lane k = neighbor rank k (nn list entry 1+k, self excluded).
// ODF partial sums = popcount(wave32 ballot & prefix mask).
// ---------------------------------------------------------------------------
__global__ __launch_bounds__(256) void point_kernel(const float* __restrict__ pts,
                                                    const float* __restrict__ col,
                                                    const float* __restrict__ V,
                                                    const int*   __restrict__ nn_idx,
                                                    float* __restrict__ odf_out,
                                                    float* __restrict__ dv_out) {
  const int lane = threadIdx.x & 31;
  const int wave = threadIdx.x >> 5;
  const int p    = blockIdx.x * 8 + wave;       // global point id, 0..B*N-1
  const int b    = p >> 12;
  const int n    = p & (NPTS - 1);
  const float* P = pts + (size_t)b * NPTS * 3;
  const float* C = col + (size_t)b * NPTS * 3;

  const float px = P[n * 3 + 0], py = P[n * 3 + 1], pz = P[n * 3 + 2];
  const int   nb = nn_idx[p * KNN + 1 + lane];  // ranks 1..32
  const float dx = P[nb * 3 + 0] - px;
  const float dy = P[nb * 3 + 1] - py;
  const float dz = P[nb * 3 + 2] - pz;

  // mean of xy offsets over the 32 neighbors (butterfly -> uniform in wave)
  float mx = dx, my = dy;
#pragma unroll
  for (int off = 16; off; off >>= 1) {
    mx += __shfl_xor(mx, off);
    my += __shfl_xor(my, off);
  }
  mx *= (1.0f / 32.0f);
  my *= (1.0f / 32.0f);
  const float nrm = sqrtf(mx * mx + my * my);
  mx /= (nrm + 1e-7f);
  my /= (nrm + 1e-7f);
  const bool keep = (mx > 0.99f);               // wave-uniform

  // Rodrigues rotation taking anchor=V[27] onto mean=(mx,my,0):
  // T v = (1 - s|c|^2) v + c x v + s c (c.v),  c = a x mean, s = 1/(1+cos)
  const float ax = V[27 * 3 + 0], ay = V[27 * 3 + 1], az = V[27 * 3 + 2];
  const float c0 = -az * my;
  const float c1 =  az * mx;
  const float c2 =  ax * my - ay * mx;
  const float cosang = mx * ax + my * ay;
  const float s  = 1.0f / (1.0f + cosang);
  const float cc = c0 * c0 + c1 * c1 + c2 * c2;
  const float k0 = 1.0f - s * cc;

  // normalized neighbor direction for this lane
  const float dn = sqrtf(dx * dx + dy * dy + dz * dz);
  const float vx = dx / dn, vy = dy / dn, vz = dz / dn;

  for (int v = 0; v < NV; ++v) {
    const float Vx = V[v * 3 + 0], Vy = V[v * 3 + 1], Vz = V[v * 3 + 2];
    float wx, wy, wz;
    if (keep) {
      wx = -Vx; wy = -Vy; wz = Vz;              // V180
    } else {
      const float cv = c0 * Vx + c1 * Vy + c2 * Vz;
      wx = k0 * Vx + (c1 * Vz - c2 * Vy) + s * c0 * cv;
      wy = k0 * Vy + (c2 * Vx - c0 * Vz) + s * c1 * cv;
      wz = k0 * Vz + (c0 * Vy - c1 * Vx) + s * c2 * cv;
    }
    const float ang = vx * wx + vy * wy + vz * wz;
    const unsigned m1 = (unsigned)__ballot(ang > 0.5f);
    const unsigned m2 = (unsigned)__ballot(ang > 0.85065057355f);
    if (lane < 8) {
      const unsigned msk = (lane < 4) ? m1 : m2;
      const int m = 8 * ((lane & 3) + 1);       // 8,16,24,32
      const unsigned pm = (m == 32) ? 0xFFFFFFFFu : ((1u << m) - 1u);
      odf_out[((size_t)p * NV + v) * 8 + lane] =
          (float)__popc(msk & pm) / (float)m;
    }
  }

  // color/point diff vectors for the 8 nearest (48 values per point)
  for (int t = lane; t < 48; t += 32) {
    const int k  = t / 6;
    const int ch = t % 6;
    const int n8 = nn_idx[p * KNN + 1 + k];
    const float self = (ch < 3) ? P[n * 3 + ch] : C[n * 3 + ch - 3];
    const float nbrv = (ch < 3) ? P[n8 * 3 + ch] : C[n8 * 3 + ch - 3];
    dv_out[((size_t)p * 8 + k) * 6 + ch] = nbrv - self;
  }
}

// ---------------------------------------------------------------------------
extern "C" void kernel_launch(void* const* d_in, const int* in_sizes, int n_in,
                              void* d_out, int out_size, void* d_ws, size_t ws_size,
                              hipStream_t stream) {
  (void)in_sizes; (void)n_in; (void)out_size; (void)ws_size;
  const float* pts = (const float*)d_in[0];
  const float* col = (const float*)d_in[1];
  const float* V   = (const float*)d_in[2];
  float* odf = (float*)d_out;                                   // (B,N,42,8)
  float* dvo = (float*)d_out + (size_t)BATCH * NPTS * NV * 8;   // (B,N,8,6)
  int*   nn  = (int*)d_ws;                                      // (B*N,33)

  const size_t shmem =
      (size_t)(ROWS * NPTS + PTS_DW + ROWS) * sizeof(float);    // ~304 KB
  knn_kernel<<<dim3(BATCH * (NPTS / ROWS)), dim3(256), shmem, stream>>>(pts, nn);
  point_kernel<<<dim3(BATCH * NPTS / 8), dim3(256), 0, stream>>>(pts, col, V, nn, odf, dvo);
}